// SequenceMergingSeq_86466281603384
// MI455X (gfx1250) — compile-verified
//
#include <hip/hip_runtime.h>
#include <cstdint>
#include <cstddef>

// Problem constants (match reference)
#define BATCH   16
#define SEQ     4096
#define DIM     128
#define NCHUNK  64
#define CHUNK   (SEQ / NCHUNK)      // 64 steps per chunk
#define SUBROWS 8                   // rows staged per TDM transfer
#define NSTAGE  (CHUNK / SUBROWS)   // 8 stages per chunk
#define WPB     2                   // waves per block
#define LN_EPS  1e-5f
#define REC_EPS 1e-8f

typedef __attribute__((ext_vector_type(4))) unsigned int u32x4_t;
typedef __attribute__((ext_vector_type(8))) int          i32x8_t;
typedef __attribute__((ext_vector_type(4))) int          i32x4_t;

// ---------------------------------------------------------------------------
// TDM: async DMA of nElems contiguous f32 from global to LDS (1-D descriptor).
// D# group0: count=1 | lds_addr | global_addr[56:0] | type=2
// D# group1: data_size=4B, tensor_dim0 = tile_dim0 = nElems, tensor_dim1=1,
//            tile_dim1/2 = 0 (unused), stride0 = nElems.
// ---------------------------------------------------------------------------
__device__ __forceinline__ void tdm_load_1d(const void* gptr, unsigned ldsByteOff,
                                            unsigned nElems)
{
  unsigned long long ga = (unsigned long long)(uintptr_t)gptr;
  u32x4_t g0;
  g0[0] = 1u;                                                   // count=1 (valid user desc)
  g0[1] = ldsByteOff;                                           // lds_addr (bytes)
  g0[2] = (unsigned)(ga & 0xFFFFFFFFull);                       // global_addr[31:0]
  g0[3] = (unsigned)((ga >> 32) & 0x01FFFFFFull) | (2u << 30);  // addr[56:32] | type=2

  i32x8_t g1;
  g1[0] = (int)(2u << 16);                    // workgroup_mask=0, data_size=2 (4 bytes)
  g1[1] = (int)((nElems & 0xFFFFu) << 16);    // tensor_dim0[15:0] @bits[63:48]
  g1[2] = (int)((nElems >> 16) | (1u << 16)); // tensor_dim0[31:16]; tensor_dim1=1
  g1[3] = (int)((nElems & 0xFFFFu) << 16);    // tile_dim0 @bits[127:112]
  g1[4] = 0;                                  // tile_dim1=0, tile_dim2=0 (unused)
  g1[5] = (int)nElems;                        // tensor_dim0_stride lo32
  g1[6] = (int)((nElems & 0xFFFFu) << 16);    // stride0 hi=0; tensor_dim1_stride lo16
  g1[7] = 0;

  i32x4_t z4 = (i32x4_t)0;
#if __clang_major__ >= 23
  i32x8_t z8 = (i32x8_t)0;
  __builtin_amdgcn_tensor_load_to_lds(g0, g1, z4, z4, z8, 0);
#else
  __builtin_amdgcn_tensor_load_to_lds(g0, g1, z4, z4, 0);
#endif
}

__device__ __forceinline__ float sigm(float x) { return 1.0f / (1.0f + __expf(-x)); }

// Intra-wave (wave32) sum-reduce of two values; no LDS, no barriers.
__device__ __forceinline__ void wred2(float& x, float& y)
{
#pragma unroll
  for (int m = 16; m >= 1; m >>= 1) {
    x += __shfl_xor(x, m, 32);
    y += __shfl_xor(y, m, 32);
  }
}

// ---------------------------------------------------------------------------
// Passes 1 & 3: per-(batch,chunk) wave scans its 64-step chunk.
// FINAL=false : start from zero state, emit chunk aggregates (P, Q, Pm, Qa).
// FINAL=true  : start from carried-in state, emit LayerNormed outputs.
// Chunk data is streamed through LDS with double-buffered TDM transfers.
// ---------------------------------------------------------------------------
template <bool FINAL>
__global__ __launch_bounds__(WPB * 32) void scan_pass(
    const float* __restrict__ C, const float* __restrict__ V, const float* __restrict__ W,
    const float* __restrict__ enc, const float* __restrict__ tmod,
    const float* __restrict__ cmod, const float* __restrict__ lnw,
    const float* __restrict__ lnb,
    float* __restrict__ Pd, float* __restrict__ Qd,
    float* __restrict__ Pm, float* __restrict__ Qa,
    const float* __restrict__ inB, const float* __restrict__ inA,
    float* __restrict__ out)
{
  extern __shared__ float smem[];

  const int lane = threadIdx.x & 31;
  const int wib  = threadIdx.x >> 5;               // wave in block
  const int wid  = blockIdx.x * WPB + wib;         // global wave-task id
  const int b    = wid / NCHUNK;
  const int k    = wid % NCHUNK;
  const int t0   = k * CHUNK;
  const int d0   = 4 * lane;                       // first of 4 channels this lane owns

  const int ARR_FLOATS  = SUBROWS * DIM;           // 1024 floats per array per stage
  const int BUF_FLOATS  = 3 * ARR_FLOATS;          // C,V,W
  const int WAVE_FLOATS = 2 * BUF_FLOATS;          // double buffered
  const int waveBase    = wib * WAVE_FLOATS;

  // Per-lane invariant vectors
  const float4 tm4  = *(const float4*)(tmod + d0);
  const float4 e4   = *(const float4*)(enc + (size_t)b * DIM + d0);
  const float4 cm4  = *(const float4*)(cmod + d0);
  float tmv[4] = {tm4.x, tm4.y, tm4.z, tm4.w};
  float ev[4]  = {e4.x, e4.y, e4.z, e4.w};
  float cmv[4] = {cm4.x, cm4.y, cm4.z, cm4.w};
  float ctx[4];
#pragma unroll
  for (int j = 0; j < 4; ++j) ctx[j] = sigm(ev[j] * cmv[j]) * ev[j];

  float lnwv[4] = {1.f, 1.f, 1.f, 1.f}, lnbv[4] = {0.f, 0.f, 0.f, 0.f};
  if (FINAL) {
    const float4 w4 = *(const float4*)(lnw + d0);
    const float4 b4 = *(const float4*)(lnb + d0);
    lnwv[0] = w4.x; lnwv[1] = w4.y; lnwv[2] = w4.z; lnwv[3] = w4.w;
    lnbv[0] = b4.x; lnbv[1] = b4.y; lnbv[2] = b4.z; lnbv[3] = b4.w;
  }

  // Recurrence state
  float a, bst[4], pvec[4], pm;
  if (FINAL) {
    const size_t cb = ((size_t)(b * NCHUNK + k)) * DIM + d0;
    const float4 ib = *(const float4*)(inB + cb);
    bst[0] = ib.x; bst[1] = ib.y; bst[2] = ib.z; bst[3] = ib.w;
    a = inA[b * NCHUNK + k];
    pm = 1.f; pvec[0] = pvec[1] = pvec[2] = pvec[3] = 1.f;  // unused
  } else {
    a = 0.f; bst[0] = bst[1] = bst[2] = bst[3] = 0.f;
    pm = 1.f; pvec[0] = pvec[1] = pvec[2] = pvec[3] = 1.f;
  }

  auto issue = [&](int stage) {
    const int buf = stage & 1;
    const size_t rowBase = ((size_t)b * SEQ + (size_t)(t0 + stage * SUBROWS)) * DIM;
    const unsigned base = (unsigned)((waveBase + buf * BUF_FLOATS) * 4);
    tdm_load_1d(C + rowBase, base + 0u * ARR_FLOATS * 4, ARR_FLOATS);
    tdm_load_1d(V + rowBase, base + 1u * ARR_FLOATS * 4, ARR_FLOATS);
    tdm_load_1d(W + rowBase, base + 2u * ARR_FLOATS * 4, ARR_FLOATS);
  };

  issue(0);

  for (int s = 0; s < NSTAGE; ++s) {
    if (s + 1 < NSTAGE) {
      issue(s + 1);
      __builtin_amdgcn_s_wait_tensorcnt(3);  // current buffer's 3 transfers done
    } else {
      __builtin_amdgcn_s_wait_tensorcnt(0);
    }
    asm volatile("" ::: "memory");

    const int   buf = s & 1;
    const float* cp = smem + waveBase + buf * BUF_FLOATS;
    const float* vp = cp + ARR_FLOATS;
    const float* wp = vp + ARR_FLOATS;

    for (int r = 0; r < SUBROWS; ++r) {
      const float4 c4 = *(const float4*)(cp + r * DIM + d0);
      const float4 v4 = *(const float4*)(vp + r * DIM + d0);
      const float4 w4 = *(const float4*)(wp + r * DIM + d0);
      float cv[4] = {c4.x, c4.y, c4.z, c4.w};
      float vv[4] = {v4.x, v4.y, v4.z, v4.w};
      float wv[4] = {w4.x, w4.y, w4.z, w4.w};

      float dec[4], ecv[4], sd = 0.f, se = 0.f;
#pragma unroll
      for (int j = 0; j < 4; ++j) {
        dec[j] = sigm(wv[j] * tmv[j]);
        ecv[j] = __expf(cv[j]);
        sd += dec[j];
        se += ecv[j];
      }
      wred2(sd, se);                       // wave-wide: sum(decay), sum(exp(c))
      const float m = sd * (1.0f / (float)DIM);

      a = m * a + se;
#pragma unroll
      for (int j = 0; j < 4; ++j) bst[j] = dec[j] * bst[j] + ecv[j] * vv[j] + ctx[j];

      if (!FINAL) {
        pm *= m;
#pragma unroll
        for (int j = 0; j < 4; ++j) pvec[j] *= dec[j];
      } else {
        const float inv = 1.0f / (a + REC_EPS);
        float res[4], ls = 0.f, lss = 0.f;
#pragma unroll
        for (int j = 0; j < 4; ++j) {
          res[j] = bst[j] * inv;
          ls += res[j];
          lss += res[j] * res[j];
        }
        wred2(ls, lss);                    // wave-wide: sum, sumsq for LayerNorm
        const float mu  = ls * (1.0f / (float)DIM);
        const float var = lss * (1.0f / (float)DIM) - mu * mu;
        const float is  = rsqrtf(var + LN_EPS);
        float o[4];
#pragma unroll
        for (int j = 0; j < 4; ++j) o[j] = (res[j] - mu) * is * lnwv[j] + lnbv[j];
        const int t = t0 + s * SUBROWS + r;
        *(float4*)(out + ((size_t)b * SEQ + t) * DIM + d0) =
            make_float4(o[0], o[1], o[2], o[3]);
      }
    }
  }

  if (!FINAL) {
    const int bk = b * NCHUNK + k;
    const size_t cb = (size_t)bk * DIM + d0;
    *(float4*)(Pd + cb) = make_float4(pvec[0], pvec[1], pvec[2], pvec[3]);
    *(float4*)(Qd + cb) = make_float4(bst[0], bst[1], bst[2], bst[3]);
    if (lane == 0) { Pm[bk] = pm; Qa[bk] = a; }
  }
}

// ---------------------------------------------------------------------------
// Pass 2: per batch, serially combine the 64 chunk aggregates into per-chunk
// carried-in states (exclusive scan over chunks). 16 waves total — trivial.
// ---------------------------------------------------------------------------
__global__ __launch_bounds__(32) void combine_pass(
    const float* __restrict__ Pd, const float* __restrict__ Qd,
    const float* __restrict__ Pm, const float* __restrict__ Qa,
    float* __restrict__ inB, float* __restrict__ inA)
{
  const int b = blockIdx.x;
  const int lane = threadIdx.x;
  const int d0 = 4 * lane;
  float c0 = 0.f, c1 = 0.f, c2 = 0.f, c3 = 0.f, ca = 0.f;
  for (int k = 0; k < NCHUNK; ++k) {
    const int bk = b * NCHUNK + k;
    const size_t base = (size_t)bk * DIM + d0;
    *(float4*)(inB + base) = make_float4(c0, c1, c2, c3);   // exclusive: state before chunk k
    if (lane == 0) inA[bk] = ca;
    const float4 p = *(const float4*)(Pd + base);
    const float4 q = *(const float4*)(Qd + base);
    c0 = p.x * c0 + q.x;
    c1 = p.y * c1 + q.y;
    c2 = p.z * c2 + q.z;
    c3 = p.w * c3 + q.w;
    ca = Pm[bk] * ca + Qa[bk];
  }
}

// ---------------------------------------------------------------------------
extern "C" void kernel_launch(void* const* d_in, const int* in_sizes, int n_in,
                              void* d_out, int out_size, void* d_ws, size_t ws_size,
                              hipStream_t stream)
{
  (void)in_sizes; (void)n_in; (void)out_size; (void)ws_size;

  const float* C    = (const float*)d_in[0];
  const float* V    = (const float*)d_in[1];
  const float* W    = (const float*)d_in[2];
  const float* enc  = (const float*)d_in[3];
  const float* tmod = (const float*)d_in[4];
  const float* cmod = (const float*)d_in[5];
  const float* lnw  = (const float*)d_in[6];
  const float* lnb  = (const float*)d_in[7];
  float* out = (float*)d_out;

  // Workspace layout (floats): P | Q | inB  (each B*NC*D) then Pm | Qa | inA
  float* ws = (float*)d_ws;
  const size_t PN = (size_t)BATCH * NCHUNK * DIM;   // 131072 floats
  float* Pd  = ws;
  float* Qd  = ws + PN;
  float* inB = ws + 2 * PN;
  float* Pm  = ws + 3 * PN;
  float* Qa  = Pm + BATCH * NCHUNK;
  float* inA = Qa + BATCH * NCHUNK;

  const size_t shmem = (size_t)WPB * 2 * 3 * SUBROWS * DIM * sizeof(float);  // 48 KB
  const dim3 grid((BATCH * NCHUNK) / WPB);   // 512 blocks
  const dim3 block(WPB * 32);                // 64 threads (2 waves)

  scan_pass<false><<<grid, block, shmem, stream>>>(
      C, V, W, enc, tmod, cmod, lnw, lnb, Pd, Qd, Pm, Qa, nullptr, nullptr, nullptr);

  combine_pass<<<BATCH, 32, 0, stream>>>(Pd, Qd, Pm, Qa, inB, inA);

  scan_pass<true><<<grid, block, shmem, stream>>>(
      C, V, W, enc, tmod, cmod, lnw, lnb, nullptr, nullptr, nullptr, nullptr, inB, inA, out);
}